// MSETLayer_45148696215795
// MI455X (gfx1250) — compile-verified
//
#include <hip/hip_runtime.h>
#include <stdint.h>

// Problem geometry (from the reference): x [16,32,224,224,3] f32,
// out [16,31,112,112,1] f32. Only the 113x113 top-left corner of each frame
// is ever read. Memory-bound: ~103 MB of HBM traffic -> ~4.4us at 23.3 TB/s.
// Data path: async global->LDS staging (ASYNCcnt), compute d in LDS with
// 4-way reuse, non-temporal output stores to keep L2 for input frames.
#define B_   16
#define T_   32
#define H_   224
#define W_   224
#define C_   3
#define OH   112
#define OW   112
#define TILE_H 16                 // output rows per block
#define DROWS  (TILE_H + 1)       // 17 rows of d needed
#define DCOLS  113                // 113 cols of d needed
#define CHUNKS_PER_ROW 85         // ceil(113*3 / 4) B128 chunks per staged row
#define ROWF   (CHUNKS_PER_ROW * 4)  // 340 floats per staged LDS row
#define ROW_STRIDE_F (W_ * C_)    // 672 floats per global row
#define FRAME_FLOATS (H_ * W_ * C_)
#define NTHREADS 256

typedef int v4i __attribute__((vector_size(16)));
typedef __attribute__((address_space(1))) v4i* gptr_v4i;
typedef __attribute__((address_space(3))) v4i* lptr_v4i;

// ---- CDNA5 async global->LDS copy (ASYNCcnt path) ----------------------
__device__ __forceinline__ void async_copy_b128(const float* g, float* l) {
#if __has_builtin(__builtin_amdgcn_global_load_async_to_lds_b128)
  __builtin_amdgcn_global_load_async_to_lds_b128(
      (gptr_v4i)(float*)g,
      (lptr_v4i)l,
      /*offset=*/0, /*cpol=*/0);
#else
  unsigned lofs = (unsigned)(uintptr_t)(__attribute__((address_space(3))) float*)l;
  asm volatile("global_load_async_to_lds_b128 %0, %1, off"
               :: "v"(lofs), "v"(g) : "memory");
#endif
}

__device__ __forceinline__ void wait_async0() {
#if __has_builtin(__builtin_amdgcn_s_wait_asynccnt)
  __builtin_amdgcn_s_wait_asynccnt(0);
#else
  asm volatile("s_wait_asynccnt 0" ::: "memory");
#endif
}

// -----------------------------------------------------------------------
__global__ __launch_bounds__(NTHREADS)
void motion_mse_kernel(const float* __restrict__ x, float* __restrict__ out) {
  // Staged input tiles for frame t and t+1 (17 rows x 340 floats each),
  // plus the channel-summed squared-diff tile d (17 x 113).
  __shared__ __align__(16) float ldsA[DROWS * ROWF];
  __shared__ __align__(16) float ldsB[DROWS * ROWF];
  __shared__ float ldsD[DROWS * DCOLS];

  const int tid = threadIdx.x;
  int bx = blockIdx.x;
  const int tileh = bx % (OH / TILE_H);  bx /= (OH / TILE_H);
  const int t     = bx % (T_ - 1);
  const int b     = bx / (T_ - 1);
  const int h0    = tileh * TILE_H;

  const float* frameA = x + (size_t)(b * T_ + t) * FRAME_FLOATS
                          + (size_t)h0 * ROW_STRIDE_F;
  const float* frameB = frameA + FRAME_FLOATS;   // frame t+1, same rows

  // ---- Stage 2 x 17 rows x 85 B128 chunks asynchronously into LDS ----
  // Global row start is 16B aligned (h*2688 bytes), chunk stride 16B;
  // chunk 84 over-reads 1 float into column 113 which exists (W=224).
  const int TOTAL_CHUNKS = 2 * DROWS * CHUNKS_PER_ROW;  // 2890
  for (int i = tid; i < TOTAL_CHUNKS; i += NTHREADS) {
    const int f  = i / (DROWS * CHUNKS_PER_ROW);
    const int j  = i - f * (DROWS * CHUNKS_PER_ROW);
    const int r  = j / CHUNKS_PER_ROW;
    const int cc = j - r * CHUNKS_PER_ROW;
    const float* g = (f ? frameB : frameA) + (size_t)r * ROW_STRIDE_F + cc * 4;
    float*       l = (f ? ldsB   : ldsA)   + r * ROWF + cc * 4;
    async_copy_b128(g, l);
  }
  wait_async0();
  __syncthreads();

  // ---- d[r][w] = sum_c (A - B)^2 over the 17 x 113 region ----
  const int ND = DROWS * DCOLS;
  for (int i = tid; i < ND; i += NTHREADS) {
    const int r = i / DCOLS;
    const int w = i - r * DCOLS;
    const int base = r * ROWF + w * 3;
    const float d0 = ldsA[base + 0] - ldsB[base + 0];
    const float d1 = ldsA[base + 1] - ldsB[base + 1];
    const float d2 = ldsA[base + 2] - ldsB[base + 2];
    ldsD[i] = d0 * d0 + d1 * d1 + d2 * d2;
  }
  __syncthreads();

  // ---- 2x2 overlapping window sum, scaled by 1/(k*k*C) = 1/12 ----
  // Output is write-once: use non-temporal stores so dead output lines
  // don't evict the reusable input frames from the 192 MB L2.
  float* o = out + ((size_t)(b * (T_ - 1) + t) * OH + h0) * OW;
  const int NO = TILE_H * OW;
  for (int i = tid; i < NO; i += NTHREADS) {
    const int r = i / OW;
    const int w = i - r * OW;
    const float s = ldsD[r * DCOLS + w]       + ldsD[r * DCOLS + w + 1]
                  + ldsD[(r + 1) * DCOLS + w] + ldsD[(r + 1) * DCOLS + w + 1];
    __builtin_nontemporal_store(s * (1.0f / 12.0f), &o[(size_t)r * OW + w]);
  }
}

extern "C" void kernel_launch(void* const* d_in, const int* in_sizes, int n_in,
                              void* d_out, int out_size, void* d_ws, size_t ws_size,
                              hipStream_t stream) {
  (void)in_sizes; (void)n_in; (void)d_ws; (void)ws_size; (void)out_size;
  const float* x = (const float*)d_in[0];
  float* out = (float*)d_out;
  const int grid = B_ * (T_ - 1) * (OH / TILE_H);  // 16*31*7 = 3472 blocks
  motion_mse_kernel<<<grid, NTHREADS, 0, stream>>>(x, out);
}